// EncoderRNN_78512002171117
// MI455X (gfx1250) — compile-verified
//
#include <hip/hip_runtime.h>
#include <math.h>

// EncoderRNN for MI455X (gfx1250): fp32 WMMA (16x16x4) end to end.
//   Kernel 1: fused embedding-gather GEMM  x_proj = emb[idx] @ W_ih^T + b_ih + b_hh
//             written directly into d_out's `outputs` region (saves 64MB ws).
//   Kernel 2: persistent-scan RNN. 128 waves own one 16x16 tile of h each,
//             W_hh tile staged in LDS, h double-buffered through L2 (d_ws),
//             per-step device-scope atomic barrier.

typedef float v2f __attribute__((ext_vector_type(2)));
typedef float v8f __attribute__((ext_vector_type(8)));

#define BB 64
#define SS 512
#define EE 512
#define HH 512
#define NBLK 32u   // blocks in scan kernel

// WMMA fp32 16x16x4 helper (8-arg VOP3P form: neg_a, A, neg_b, B, c_mod, C, reuse_a, reuse_b)
__device__ __forceinline__ v8f wmma_f32(v2f a, v2f b, v8f c) {
    return __builtin_amdgcn_wmma_f32_16x16x4_f32(false, a, false, b, (short)0, c, false, false);
}

// ---------------- init: zero h0 and the barrier counters (deterministic per launch) ----
__global__ void rnn_init_ws(float* hb0, unsigned int* cnt) {
    int i = blockIdx.x * blockDim.x + threadIdx.x;
    if (i < BB * HH) hb0[i] = 0.0f;
    if (i < SS) cnt[i] = 0u;
}

// ---------------- Kernel 1: gather + projection GEMM -----------------------------------
// M = B*S = 32768, N = H = 512, K = E = 512.
// Each wave computes a 32(M) x 64(N) block: 2 A-tiles x 4 B-tiles -> 8 v8f accumulators.
// A layout (16x4 f32): lane = M%16 + 16*(K-half); v0 holds K=+0/+2, v1 holds K=+1/+3.
// B layout (4x16 f32): mirrored -> both A and B load as one float2 per lane per step.
__global__ __launch_bounds__(256) void xproj_gemm(
    const long long* __restrict__ idx, const float* __restrict__ emb,
    const float* __restrict__ W_ih,   const float* __restrict__ b_ih,
    const float* __restrict__ b_hh,   float* __restrict__ out)
{
    const int lane   = threadIdx.x & 31;
    const int lane16 = lane & 15;
    const int half   = lane >> 4;
    const int w      = blockIdx.x * 8 + (threadIdx.x >> 5);
    const int mBase  = (w >> 3) * 32;      // 1024 m-tiles of 32 rows
    const int nBase  = (w & 7) * 64;       // 8 n-tiles of 64 cols

    const float* arow0 = emb + (size_t)idx[mBase + lane16] * EE;
    const float* arow1 = emb + (size_t)idx[mBase + 16 + lane16] * EE;
    const float* brow[4];
#pragma unroll
    for (int j = 0; j < 4; ++j)
        brow[j] = W_ih + (size_t)(nBase + j * 16 + lane16) * EE;

    v8f acc[2][4];
#pragma unroll
    for (int i = 0; i < 2; ++i)
#pragma unroll
        for (int j = 0; j < 4; ++j) acc[i][j] = {};

    const int koff = 2 * half;
#pragma unroll 2
    for (int k0 = 0; k0 < EE; k0 += 4) {
        v2f a0 = *(const v2f*)(arow0 + k0 + koff);
        v2f a1 = *(const v2f*)(arow1 + k0 + koff);
#pragma unroll
        for (int j = 0; j < 4; ++j) {
            v2f b = *(const v2f*)(brow[j] + k0 + koff);
            acc[0][j] = wmma_f32(a0, b, acc[0][j]);
            acc[1][j] = wmma_f32(a1, b, acc[1][j]);
        }
    }

    // C/D layout: reg r, lanes0-15 -> M=r, lanes16-31 -> M=r+8; N = lane16. Bias is per-column.
#pragma unroll
    for (int j = 0; j < 4; ++j) {
        const int n = nBase + j * 16 + lane16;
        const float bias = b_ih[n] + b_hh[n];
#pragma unroll
        for (int i = 0; i < 2; ++i) {
#pragma unroll
            for (int r = 0; r < 8; ++r) {
                const int m = mBase + i * 16 + r + 8 * half;   // m = b*S + t
                out[(size_t)m * HH + n] = acc[i][j][r] + bias;
            }
        }
    }
}

// ---------------- Kernel 2: sequential scan --------------------------------------------
// 32 blocks x 128 threads (4 waves). Wave (blockIdx=nt, waveId=mt) owns h-tile
// rows [16*mt,16*mt+16) x cols [16*nt,16*nt+16). Its W_hh slice lives in LDS.
__global__ __launch_bounds__(128) void rnn_scan(
    const float* __restrict__ W_hh, float* __restrict__ out,
    float* __restrict__ hb0, float* __restrict__ hb1, unsigned int* __restrict__ cnt)
{
    __shared__ float ldsB[16 * 514];               // 16 rows of W_hh, padded (+2) vs 64 banks

    const int lane   = threadIdx.x & 31;
    const int lane16 = lane & 15;
    const int half   = lane >> 4;
    const int mt     = threadIdx.x >> 5;           // 0..3  (batch tile)
    const int nt     = blockIdx.x;                 // 0..31 (hidden tile)
    const int mBase  = mt * 16;
    const int nBase  = nt * 16;

    for (int i = threadIdx.x; i < 16 * HH; i += blockDim.x) {
        const int r = i >> 9, k = i & 511;
        ldsB[r * 514 + k] = W_hh[(size_t)(nBase + r) * HH + k];
    }
    __syncthreads();

    const int koff = 2 * half;
    float* hbuf[2] = { hb0, hb1 };
    float* hidden  = out + (size_t)BB * SS * HH;

    for (int t = 0; t < SS; ++t) {
        const float* hc = hbuf[t & 1];
        float*       hn = hbuf[(t + 1) & 1];

        v8f acc = {};
        const float* arow  = hc + (size_t)(mBase + lane16) * HH;
        const float* bbase = &ldsB[lane16 * 514];
#pragma unroll 4
        for (int k0 = 0; k0 < HH; k0 += 4) {
            v2f a = *(const v2f*)(arow + k0 + koff);
            v2f b = *(const v2f*)(bbase + k0 + koff);
            acc = wmma_f32(a, b, acc);
        }

#pragma unroll
        for (int r = 0; r < 8; ++r) {
            const int brow2 = mBase + r + 8 * half;            // batch index
            const int n     = nBase + lane16;
            const size_t o  = ((size_t)brow2 * SS + t) * HH + n;
            const float hv  = tanhf(acc[r] + out[o]);          // out[o] holds x_proj tile
            out[o] = hv;                                       // overwrite with final output
            hn[(size_t)brow2 * HH + n] = hv;                   // next-step h
            if (t == SS - 1) hidden[(size_t)brow2 * HH + n] = hv;
        }

        // device-wide step barrier: publish h, arrive, spin, acquire
        __threadfence();
        __syncthreads();
        if (threadIdx.x == 0) {
            __hip_atomic_fetch_add(&cnt[t], 1u, __ATOMIC_ACQ_REL, __HIP_MEMORY_SCOPE_AGENT);
            while (__hip_atomic_load(&cnt[t], __ATOMIC_ACQUIRE, __HIP_MEMORY_SCOPE_AGENT) < NBLK) {
                __builtin_amdgcn_s_sleep(1);
            }
        }
        __syncthreads();
    }
}

extern "C" void kernel_launch(void* const* d_in, const int* in_sizes, int n_in,
                              void* d_out, int out_size, void* d_ws, size_t ws_size,
                              hipStream_t stream)
{
    const long long* idx  = (const long long*)d_in[0];  // int64 token ids (B,S)
    const float*     emb  = (const float*)d_in[1];      // (V,E)
    const float*     W_ih = (const float*)d_in[2];      // (H,E)
    const float*     W_hh = (const float*)d_in[3];      // (H,H)
    const float*     b_ih = (const float*)d_in[4];      // (H,)
    const float*     b_hh = (const float*)d_in[5];      // (H,)
    float* out = (float*)d_out;                         // outputs (B,S,H) then hidden (1,B,H)

    float* hb0        = (float*)d_ws;                   // h double buffers: 2 * 128KB
    float* hb1        = hb0 + BB * HH;
    unsigned int* cnt = (unsigned int*)(hb1 + BB * HH); // 512 step counters

    rnn_init_ws<<<(BB * HH + 255) / 256, 256, 0, stream>>>(hb0, cnt);
    xproj_gemm<<<1024, 256, 0, stream>>>(idx, emb, W_ih, b_ih, b_hh, out);
    rnn_scan<<<NBLK, 128, 0, stream>>>(W_hh, out, hb0, hb1, cnt);
}